// PositionEmbeddingSine_75496935129789
// MI455X (gfx1250) — compile-verified
//
#include <hip/hip_runtime.h>
#include <stdint.h>

// ---------------------------------------------------------------------------
// PositionEmbeddingSine for MI455X (gfx1250, wave32)
//
// Roofline: output = 16*128*256*256 f32 = 512 MiB streamed once
//           => ~23 us floor at 23.3 TB/s HBM. Everything else is noise.
// Design:   phase 1 computes normalized cumsums (x_embed / y_embed) with
//           V_WMMA_F32_16X16X4_F32-based 256-element scans (exact: inputs are
//           0/1, sums <= 256). Phase 2 streams sin/cos with v_sin_f32 /
//           v_cos_f32 (hardware takes revolutions: sin(p*k*pi)=v_sin(p*k/2))
//           and 128-bit non-temporal stores (512 MiB > 192 MB L2).
// Workspace: 2 * [16,256,256] f32 = 8 MiB in d_ws.
// ---------------------------------------------------------------------------

typedef float v2f __attribute__((ext_vector_type(2)));
typedef float v4f __attribute__((ext_vector_type(4)));
typedef float v8f __attribute__((ext_vector_type(8)));

#define BATCH 16
#define H_ 256
#define W_ 256
#define NPIX (BATCH * H_ * W_) // 1048576

// ---------------- Phase 1: normalized prefix sums via WMMA -----------------
// One wave scans one 256-vector (a mask row or a mask column).
// Scan of v reshaped to R[16][16]:  T = R x U  (U upper-tri ones, incl diag)
// gives within-segment inclusive scans; segment offsets are an exclusive scan
// of T[:,15]. Computed with 4 chained V_WMMA_F32_16X16X4_F32 (K = 16).
__global__ __launch_bounds__(256) void pes_scan_kernel(
    const unsigned char* __restrict__ mask,
    float* __restrict__ xn, float* __restrict__ yn) {
  __shared__ float stage[8][256];    // per-wave input vector
  __shared__ float tbuf[8][16][16];  // per-wave T = R x U (row major)
  __shared__ float offs[8][16];      // per-wave segment offsets

  const int tid  = threadIdx.x;
  const int w    = tid >> 5;   // wave in block
  const int lane = tid & 31;
  const int m    = lane & 15;  // M row (A) / N col (B,D)
  const int hi   = lane >> 4;  // lane half select

  const bool rowMode = blockIdx.x < 512;
  const int vec = (rowMode ? (int)blockIdx.x : (int)blockIdx.x - 512) * 8 + w;
  const int b   = vec >> 8;
  const int rc  = vec & 255;   // row i (rowMode) or column j (colMode)

  // Load not_mask 256-vector into LDS (coalesced in row mode).
#pragma unroll
  for (int q = 0; q < 8; ++q) {
    int p = q * 32 + lane;
    unsigned char mv = rowMode ? mask[(b * H_ + rc) * W_ + p]
                               : mask[(b * H_ + p) * W_ + rc];
    stage[w][p] = mv ? 0.0f : 1.0f;
  }
  __syncthreads();

  // T = R x U, chunked over K in 4s. f32 A layout (ISA 7.12.2):
  //   VGPR0: K=0 (lanes 0-15) / K=2 (lanes 16-31); VGPR1: K=1 / K=3.
  // B layout assumed mirrored (row K striped across lanes within a VGPR half).
  v8f acc = {0.f, 0.f, 0.f, 0.f, 0.f, 0.f, 0.f, 0.f};
#pragma unroll
  for (int c = 0; c < 4; ++c) {
    const int K0 = 4 * c + 2 * hi;
    v2f a, bm;
    a.x  = stage[w][16 * m + K0];
    a.y  = stage[w][16 * m + K0 + 1];
    bm.x = (K0 <= m) ? 1.0f : 0.0f;      // U[K0][n], n = m
    bm.y = (K0 + 1 <= m) ? 1.0f : 0.0f;  // U[K0+1][n]
    acc = __builtin_amdgcn_wmma_f32_16x16x4_f32(false, a, false, bm,
                                                (short)0, acc, false, false);
  }

  // Scatter D (lane<16: VGPR r -> (M=r, N=lane); lane>=16: M=r+8) to LDS.
#pragma unroll
  for (int r = 0; r < 8; ++r) tbuf[w][r + 8 * hi][m] = acc[r];
  __syncthreads();

  // Exclusive scan of segment totals T[s][15] (16 values, trivial).
  if (lane < 16) {
    float o = 0.0f;
    for (int s = 0; s < lane; ++s) o += tbuf[w][s][15];
    offs[w][lane] = o;
  }
  __syncthreads();

  const float last = tbuf[w][15][15] + offs[w][15];
  const float inv  = 1.0f / (last + 1e-6f);

#pragma unroll
  for (int q = 0; q < 8; ++q) {
    int p = q * 32 + lane;
    float cum = tbuf[w][p >> 4][p & 15] + offs[w][p >> 4];
    float val = (cum - 0.5f) * inv;
    if (rowMode) xn[(b * H_ + rc) * W_ + p] = val;  // x_embed (normalized)
    else         yn[(b * H_ + p) * W_ + rc] = val;  // y_embed (normalized)
  }
}

// ---------------- Phase 2: stream sin/cos output (bandwidth-bound) ---------
// Block = (b, i). Thread owns a j-quad and one channel residue; 16 iterations
// cover sin channels 0..63, each also emitting the matching cos channel c+64.
__global__ __launch_bounds__(256) void pes_emit_kernel(
    const float* __restrict__ xn, const float* __restrict__ yn,
    float* __restrict__ out) {
  const int blk = blockIdx.x;  // 0..4095
  const int b = blk >> 8;
  const int i = blk & 255;
  const int t = threadIdx.x;
  const int jq = (t & 63) * 4;  // j quad start
  const int cr = t >> 6;        // channel residue 0..3

  const size_t rowoff = (size_t)(b * H_ + i) * W_ + jq;
  const v4f xq = *(const v4f*)(xn + rowoff);
  const v4f yq = *(const v4f*)(yn + rowoff);

#pragma unroll 4
  for (int it = 0; it < 16; ++it) {
    const int c = it * 4 + cr;  // 0..63 (sin channel; cos channel = c+64)
    const int d = c >> 1;
    // angle in revolutions: pos*(d+1)*pi / (2*pi) = pos*(d+1)*0.5
    const float fac = 0.5f * (float)(d + 1);
    const v4f pos = (c & 1) ? yq : xq;
    const v4f ang = pos * fac;
    v4f s, co;
#pragma unroll
    for (int e = 0; e < 4; ++e) {
      s[e]  = __builtin_amdgcn_sinf(ang[e]);  // v_sin_f32 (revolutions)
      co[e] = __builtin_amdgcn_cosf(ang[e]);  // v_cos_f32
    }
    const size_t base = (((size_t)(b * 128 + c) * H_) + i) * W_ + jq;
    __builtin_nontemporal_store(s, (v4f*)(out + base));
    __builtin_nontemporal_store(co, (v4f*)(out + base + (size_t)64 * H_ * W_));
  }
}

extern "C" void kernel_launch(void* const* d_in, const int* in_sizes, int n_in,
                              void* d_out, int out_size, void* d_ws,
                              size_t ws_size, hipStream_t stream) {
  (void)in_sizes; (void)n_in; (void)out_size; (void)ws_size;
  const unsigned char* mask = (const unsigned char*)d_in[0];  // bool mask
  float* xn = (float*)d_ws;       // [16,256,256] normalized x_embed (4 MiB)
  float* yn = xn + NPIX;          // [16,256,256] normalized y_embed (4 MiB)
  float* out = (float*)d_out;     // [16,128,256,256]

  // 512 row-scan blocks + 512 column-scan blocks, 8 waves each (1 scan/wave).
  pes_scan_kernel<<<1024, 256, 0, stream>>>(mask, xn, yn);
  // 4096 blocks, each streams 128 KiB of output.
  pes_emit_kernel<<<4096, 256, 0, stream>>>(xn, yn, out);
}